// Cond_TBC_85495618994674
// MI455X (gfx1250) — compile-verified
//
#include <hip/hip_runtime.h>
#include <math.h>
#include <stdint.h>

typedef float v2f __attribute__((ext_vector_type(2)));
typedef float v8f __attribute__((ext_vector_type(8)));

#define CIN   256
#define COUT  256
#define HH    56
#define WW    56
#define KC    16            // cin channels per LDS chunk
#define KS    20            // k-stride (floats) in LDS: even, 16B-friendly, conflict-free
#define RB    8             // output rows per block
#define XROWS (RB + 2)      // staged input rows (halo)
#define XW    66            // padded width: w = -1..64
#define EXP_STRIDE 589824   // 256*256*9 floats per expert
#define AGG_ELEMS (32 * 9 * 256 * 256)  // mixed weights, [b][rs][co][ci]

// ---------------- Kernel 1: global average pool -> pooled[b*CIN + c] ----------------
__global__ __launch_bounds__(256) void pool_kernel(const float* __restrict__ x,
                                                   float* __restrict__ pooled) {
    int bc = blockIdx.x;
    const float* p = x + (size_t)bc * (HH * WW);
    float s = 0.f;
    for (int i = threadIdx.x; i < HH * WW; i += 256) s += p[i];
    __shared__ float red[256];
    red[threadIdx.x] = s;
    __syncthreads();
    for (int off = 128; off > 0; off >>= 1) {
        if (threadIdx.x < off) red[threadIdx.x] += red[threadIdx.x + off];
        __syncthreads();
    }
    if (threadIdx.x == 0) pooled[bc] = red[0] * (1.0f / (HH * WW));
}

// ---------------- Kernel 2: att[b,k] = sigmoid(pooled[b,:] . att_w[k,:]) ----------------
__global__ __launch_bounds__(128) void att_kernel(const float* __restrict__ pooled,
                                                  const float* __restrict__ att_w,
                                                  float* __restrict__ att) {
    int t = threadIdx.x;
    if (t < 32 * 4) {
        int b = t >> 2, k = t & 3;
        float s = 0.f;
        for (int c = 0; c < CIN; ++c) s += pooled[b * CIN + c] * att_w[k * CIN + c];
        att[t] = 1.0f / (1.0f + expf(-s));
    }
}

// ---------------- Kernel 3: premix expert weights -> aggW[b][rs][co][ci] ----------------
__global__ __launch_bounds__(256) void premix_kernel(const float* __restrict__ weight,
                                                     const float* __restrict__ att,
                                                     float* __restrict__ aggW) {
    int idx = blockIdx.x * 256 + threadIdx.x;
    if (idx >= AGG_ELEMS) return;
    int ci = idx & 255;
    int co = (idx >> 8) & 255;
    int t  = idx >> 16;            // b*9 + rs
    int rs = t % 9;
    int b  = t / 9;
    size_t off = ((size_t)co * CIN + ci) * 9 + rs;
    float v = att[b * 4 + 0] * weight[off]
            + att[b * 4 + 1] * weight[off + 1 * EXP_STRIDE]
            + att[b * 4 + 2] * weight[off + 2 * EXP_STRIDE]
            + att[b * 4 + 3] * weight[off + 3 * EXP_STRIDE];
    aggW[idx] = v;
}

// ---------------- Kernel 4: implicit-GEMM 3x3 conv, f32 WMMA, async LDS staging ----------
// Block: 256 thr = 8 waves (4 waveM x 2 waveN). Tile: M=64 cout, N = 8 rows x 56 w.
// Grid: (COUT/64, HH/RB, B) = (4, 7, 32).
template <bool PREMIX>
__global__ __launch_bounds__(256) void condconv_kernel(
    const float* __restrict__ x,
    const float* __restrict__ wsrc,   // PREMIX: aggW[b][rs][co][ci]; else raw weight[k][co][ci][rs]
    const float* __restrict__ att,    // only used when !PREMIX
    float* __restrict__ out) {
    __shared__ __align__(16) float Xlds[XROWS * XW * KS];   // 52.8 KB  [ri][iw][k]
    __shared__ __align__(16) float Wlds[9 * 64 * KS];       // 46.1 KB  [rs][m][k]

    const int coBase = blockIdx.x * 64;
    const int hBase  = blockIdx.y * RB;
    const int b      = blockIdx.z;

    const int tid   = threadIdx.x;
    const int lane  = tid & 31;
    const int wave  = tid >> 5;
    const int waveM = wave & 3;
    const int waveN = wave >> 2;
    const int lmod  = lane & 15;
    const int kHi   = (lane >> 4) * 2;         // fragment K sub-offset per lane half
    const int mA    = waveM * 16 + lmod;       // A row (cout within tile)

    float a0 = 0.f, a1 = 0.f, a2 = 0.f, a3 = 0.f;
    if (!PREMIX) {
        a0 = att[b * 4 + 0]; a1 = att[b * 4 + 1];
        a2 = att[b * 4 + 2]; a3 = att[b * 4 + 3];
    }

    v8f acc[RB][2];
    #pragma unroll
    for (int i = 0; i < RB; ++i) { acc[i][0] = (v8f){}; acc[i][1] = (v8f){}; }

    for (int ci0 = 0; ci0 < CIN; ci0 += KC) {
        // ---- stage weight tile: Wlds[rs][m][k] ----
        if (PREMIX) {
            // pure copy -> async b128 (16B packets), 2304 packets
            for (int i = tid; i < 9 * 64 * 4; i += 256) {
                int q  = i & 3;
                int m  = (i >> 2) & 63;
                int rs = i >> 8;
                unsigned lds  = (unsigned)(uintptr_t)&Wlds[(rs * 64 + m) * KS + q * 4];
                unsigned voff = (unsigned)(((((b * 9 + rs) * COUT) + coBase + m) * CIN
                                            + ci0 + q * 4) * 4);
                asm volatile("global_load_async_to_lds_b128 %0, %1, %2"
                             :: "v"(lds), "v"(voff), "s"(wsrc) : "memory");
            }
        } else {
            // mix 4 experts on the fly (L2-resident weights)
            for (int i = tid; i < 9 * 64 * KC; i += 256) {
                int kk = i & 15;
                int m  = (i >> 4) & 63;
                int rs = i >> 10;
                size_t off = ((size_t)(coBase + m) * CIN + ci0 + kk) * 9 + rs;
                float v = a0 * wsrc[off]
                        + a1 * wsrc[off + 1 * EXP_STRIDE]
                        + a2 * wsrc[off + 2 * EXP_STRIDE]
                        + a3 * wsrc[off + 3 * EXP_STRIDE];
                Wlds[(rs * 64 + m) * KS + kk] = v;
            }
        }
        // ---- stage x slab: Xlds[ri][iw][k], async b32, zero-fill halo OOB ----
        for (int i = tid; i < KC * XROWS * XW; i += 256) {
            int k   = i / (XROWS * XW);
            int rem = i - k * (XROWS * XW);
            int ri  = rem / XW;
            int iw  = rem - ri * XW;
            int hi  = hBase + ri - 1;
            int wi  = iw - 1;
            int idx = (ri * XW + iw) * KS + k;
            if ((unsigned)hi < HH && (unsigned)wi < WW) {
                unsigned lds  = (unsigned)(uintptr_t)&Xlds[idx];
                unsigned voff = (unsigned)(((((b * CIN) + ci0 + k) * HH + hi) * WW + wi) * 4);
                asm volatile("global_load_async_to_lds_b32 %0, %1, %2"
                             :: "v"(lds), "v"(voff), "s"(x) : "memory");
            } else {
                Xlds[idx] = 0.f;
            }
        }
        asm volatile("s_wait_asynccnt 0" ::: "memory");
        __syncthreads();

        // ---- compute: per kc, 9 A-frags (b64) + shared B-frags, 144 WMMA ----
        const int nb0 = waveN * 32 + lmod;
        #pragma unroll 1
        for (int kc = 0; kc < KC; kc += 4) {
            const int kk = kc + kHi;
            v2f A[9];
            #pragma unroll
            for (int rs = 0; rs < 9; ++rs)
                A[rs] = *(const v2f*)&Wlds[(rs * 64 + mA) * KS + kk];
            #pragma unroll
            for (int s = 0; s < 3; ++s) {
                #pragma unroll
                for (int ri = 0; ri < XROWS; ++ri) {
                    v2f B0 = *(const v2f*)&Xlds[(ri * XW + nb0 + s) * KS + kk];
                    v2f B1 = *(const v2f*)&Xlds[(ri * XW + nb0 + 16 + s) * KS + kk];
                    #pragma unroll
                    for (int r = 0; r < 3; ++r) {
                        const int hp = ri - r;          // compile-time after unroll
                        if (hp >= 0 && hp < RB) {
                            acc[hp][0] = __builtin_amdgcn_wmma_f32_16x16x4_f32(
                                false, A[r * 3 + s], false, B0, (short)0, acc[hp][0], false, false);
                            acc[hp][1] = __builtin_amdgcn_wmma_f32_16x16x4_f32(
                                false, A[r * 3 + s], false, B1, (short)0, acc[hp][1], false, false);
                        }
                    }
                }
            }
        }
        __syncthreads();
    }

    // ---- store: D layout: VGPR v -> M = v + 8*(lane>>4); N = lane&15 ----
    const int w0   = waveN * 32 + lmod;
    const int mRow = coBase + waveM * 16 + (lane >> 4) * 8;
    #pragma unroll
    for (int hp = 0; hp < RB; ++hp) {
        #pragma unroll
        for (int v = 0; v < 8; ++v) {
            size_t o = (((size_t)b * COUT + (mRow + v)) * HH + (hBase + hp)) * WW;
            if (w0 < WW)      out[o + w0]      = acc[hp][0][v];
            if (w0 + 16 < WW) out[o + w0 + 16] = acc[hp][1][v];
        }
    }
}

extern "C" void kernel_launch(void* const* d_in, const int* in_sizes, int n_in,
                              void* d_out, int out_size, void* d_ws, size_t ws_size,
                              hipStream_t stream) {
    const float* x      = (const float*)d_in[0];   // (32,256,56,56)
    const float* att_w  = (const float*)d_in[1];   // (4,256)
    const float* weight = (const float*)d_in[2];   // (4,256,256,3,3)
    float* out = (float*)d_out;

    float* pooled = (float*)d_ws;                  // 8192 floats
    float* att    = pooled + 32 * CIN;             // 128 floats
    float* aggW   = pooled + 8320;                 // 16B-aligned, AGG_ELEMS floats

    pool_kernel<<<32 * CIN, 256, 0, stream>>>(x, pooled);
    att_kernel<<<1, 128, 0, stream>>>(pooled, att_w, att);

    dim3 grid(COUT / 64, HH / RB, 32);
    const size_t need = (size_t)(8320 + (size_t)AGG_ELEMS) * sizeof(float);
    if (ws_size >= need) {
        premix_kernel<<<(AGG_ELEMS + 255) / 256, 256, 0, stream>>>(weight, att, aggW);
        condconv_kernel<true><<<grid, 256, 0, stream>>>(x, aggW, att, out);
    } else {
        condconv_kernel<false><<<grid, 256, 0, stream>>>(x, weight, att, out);
    }
}